// SGC_80977313399737
// MI455X (gfx1250) — compile-verified
//
#include <hip/hip_runtime.h>
#include <hip/hip_bf16.h>

typedef float v2f __attribute__((ext_vector_type(2)));
typedef float v8f __attribute__((ext_vector_type(8)));

#define CH 512

// ---------------- degree / norm ----------------
__global__ void init_deg_kernel(float* d1, float* d2, int N) {
  int i = blockIdx.x * blockDim.x + threadIdx.x;
  if (i < N) { d1[i] = 1.0f; d2[i] = 1.0f; }   // self-loop contributes 1
}

__global__ void count_deg_kernel(const int* __restrict__ cols, float* deg, int E) {
  int e = blockIdx.x * blockDim.x + threadIdx.x;
  if (e < E) atomicAdd(&deg[cols[e]], 1.0f);
}

__global__ void dinv_kernel(float* deg, int N) {
  int i = blockIdx.x * blockDim.x + threadIdx.x;
  if (i < N) deg[i] = 1.0f / sqrtf(deg[i]);    // deg >= 1 always
}

// ---------------- propagation ----------------
// out[i][:] = h[i][:] * dinv[i]^2   (self-loop term; also zero-initializes out)
__global__ void self_scale_kernel(const float* __restrict__ h, float* __restrict__ out,
                                  const float* __restrict__ dinv, int nvec) {
  int i = blockIdx.x * blockDim.x + threadIdx.x;
  if (i >= nvec) return;
  int node = i >> 7;                  // 128 float4 per node (512 ch)
  float s = dinv[node]; s *= s;
  float4 v = ((const float4*)h)[i];
  v.x *= s; v.y *= s; v.z *= s; v.w *= s;
  ((float4*)out)[i] = v;
}

// out[col][:] += h[row][:] * dinv[row]*dinv[col]; 128 lanes per edge, float4 each
__global__ void edge_scatter_kernel(const float* __restrict__ h, float* __restrict__ out,
                                    const int* __restrict__ rows, const int* __restrict__ cols,
                                    const float* __restrict__ dinv, int E) {
  int e = (int)(blockIdx.x * 2 + (threadIdx.x >> 7));
  if (e >= E) return;
  int t = threadIdx.x & 127;
  int r = rows[e], c = cols[e];
  float norm = dinv[r] * dinv[c];
  float4 v = *(const float4*)(h + (size_t)r * CH + t * 4);
  float* o = out + (size_t)c * CH + t * 4;
  atomicAdd(o + 0, v.x * norm);
  atomicAdd(o + 1, v.y * norm);
  atomicAdd(o + 2, v.z * norm);
  atomicAdd(o + 3, v.w * norm);
}

// ---------------- fp32 WMMA GEMM: C[M,512] = H[M,512] @ W[512,512] + bias ----------------
// Each wave computes a 16x128 strip of C (8 accumulator tiles), K-loop step 4.
__global__ void __launch_bounds__(256)
gemm_wmma_kernel(const float* __restrict__ H, const float* __restrict__ Wm,
                 const float* __restrict__ bias, float* __restrict__ C, int M) {
  const int K = CH, N = CH;
  int wave = threadIdx.x >> 5;
  int lane = threadIdx.x & 31;
  int lr   = lane & 15;
  int hi   = lane >> 4;
  int row0 = blockIdx.x * 128 + wave * 16;   // wave-uniform
  if (row0 >= M) return;                     // uniform exit -> EXEC all-1s for WMMA
  int col0 = blockIdx.y * 128;

  v8f acc[8] = {};

  // A frag: lane holds A[row0+lr][k0 + 2*hi + {0,1}]
  const float* aptr = H + (size_t)(row0 + lr) * K + 2 * hi;
  // B frag: lane holds W[k0 + 2*hi + {0,1}][col0 + t*16 + lr]
  const float* bptr = Wm + (size_t)(2 * hi) * N + col0 + lr;

  for (int k0 = 0; k0 < K; k0 += 4) {
    v2f a;
    a.x = aptr[0];
    a.y = aptr[1];
#pragma unroll
    for (int t = 0; t < 8; t++) {
      v2f b;
      b.x = bptr[t * 16];
      b.y = bptr[t * 16 + N];
      acc[t] = __builtin_amdgcn_wmma_f32_16x16x4_f32(
          false, a, false, b, (short)0, acc[t], false, false);
    }
    aptr += 4;
    bptr += (size_t)4 * N;
  }

#pragma unroll
  for (int t = 0; t < 8; t++) {
    int n = col0 + t * 16 + lr;
    float bv = bias[n];
#pragma unroll
    for (int v = 0; v < 8; v++) {
      int r = row0 + v + 8 * hi;             // C layout: VGPR v -> rows v / v+8
      C[(size_t)r * N + n] = acc[t][v] + bv;
    }
  }
}

// ---------------- softmax (+argmax) per 512-wide row, in place ----------------
__global__ void __launch_bounds__(128)
softmax_argmax_kernel(const float* __restrict__ in, float* __restrict__ out,
                      float* __restrict__ pred, int M) {
  __shared__ float sval[128];
  __shared__ int   sidx[128];
  int row = blockIdx.x;
  if (row >= M) return;
  int t = threadIdx.x;
  float4 v = *(const float4*)(in + (size_t)row * CH + t * 4);

  float m = v.x; int mi = t * 4;
  if (v.y > m) { m = v.y; mi = t * 4 + 1; }
  if (v.z > m) { m = v.z; mi = t * 4 + 2; }
  if (v.w > m) { m = v.w; mi = t * 4 + 3; }
  sval[t] = m; sidx[t] = mi;
  __syncthreads();
  for (int s = 64; s > 0; s >>= 1) {
    if (t < s) {
      float o = sval[t + s]; int oi = sidx[t + s];
      if (o > sval[t] || (o == sval[t] && oi < sidx[t])) { sval[t] = o; sidx[t] = oi; }
    }
    __syncthreads();
  }
  float rowmax = sval[0];
  int   rowidx = sidx[0];
  __syncthreads();

  float4 e;
  e.x = expf(v.x - rowmax);
  e.y = expf(v.y - rowmax);
  e.z = expf(v.z - rowmax);
  e.w = expf(v.w - rowmax);
  sval[t] = e.x + e.y + e.z + e.w;
  __syncthreads();
  for (int s = 64; s > 0; s >>= 1) {
    if (t < s) sval[t] += sval[t + s];
    __syncthreads();
  }
  float inv = 1.0f / sval[0];
  float4 o4; o4.x = e.x * inv; o4.y = e.y * inv; o4.z = e.z * inv; o4.w = e.w * inv;
  *(float4*)(out + (size_t)row * CH + t * 4) = o4;
  if (pred != nullptr && t == 0) pred[row] = (float)rowidx;
}

// ---------------- host ----------------
extern "C" void kernel_launch(void* const* d_in, const int* in_sizes, int n_in,
                              void* d_out, int out_size, void* d_ws, size_t ws_size,
                              hipStream_t stream) {
  const float* x  = (const float*)d_in[0];
  const int*   e1 = (const int*)d_in[1];
  const int*   e2 = (const int*)d_in[2];
  const float* W1 = (const float*)d_in[3];
  const float* b1 = (const float*)d_in[4];
  const float* W2 = (const float*)d_in[5];
  const float* b2 = (const float*)d_in[6];

  const int N  = in_sizes[0] / CH;   // 50000 nodes
  const int E  = in_sizes[1] / 2;    // 800000 edges
  const int E2 = in_sizes[2] / 2;

  // workspace: deg arrays + two N*512 fp32 ping-pong buffers
  float* deg1 = (float*)d_ws;
  float* deg2 = deg1 + N;
  size_t off  = ((size_t)2 * N * sizeof(float) + 255) & ~(size_t)255;
  float* bufA = (float*)((char*)d_ws + off);
  float* bufB = bufA + (size_t)N * CH;

  float* logits1 = (float*)d_out;
  float* logits2 = logits1 + (size_t)N * CH;
  float* preds   = logits2 + (size_t)N * CH;

  // degrees (deg over destination/col, self-loop included via init=1)
  init_deg_kernel<<<(N + 255) / 256, 256, 0, stream>>>(deg1, deg2, N);
  count_deg_kernel<<<(E + 255) / 256, 256, 0, stream>>>(e1 + E, deg1, E);
  count_deg_kernel<<<(E2 + 255) / 256, 256, 0, stream>>>(e2 + E2, deg2, E2);
  dinv_kernel<<<(N + 255) / 256, 256, 0, stream>>>(deg1, N);
  dinv_kernel<<<(N + 255) / 256, 256, 0, stream>>>(deg2, N);

  const int nvec = N * (CH / 4);
  dim3 gscale((nvec + 255) / 256);
  dim3 gedge1((E + 1) / 2), gedge2((E2 + 1) / 2);
  dim3 ggemm((N + 127) / 128, CH / 128);

  // conv 1: two propagation rounds, GEMM, softmax+argmax
  self_scale_kernel<<<gscale, 256, 0, stream>>>(x, bufA, deg1, nvec);
  edge_scatter_kernel<<<gedge1, 256, 0, stream>>>(x, bufA, e1, e1 + E, deg1, E);
  self_scale_kernel<<<gscale, 256, 0, stream>>>(bufA, bufB, deg1, nvec);
  edge_scatter_kernel<<<gedge1, 256, 0, stream>>>(bufA, bufB, e1, e1 + E, deg1, E);
  gemm_wmma_kernel<<<ggemm, 256, 0, stream>>>(bufB, W1, b1, logits1, N);
  softmax_argmax_kernel<<<N, 128, 0, stream>>>(logits1, logits1, preds, N);

  // conv 2
  self_scale_kernel<<<gscale, 256, 0, stream>>>(x, bufA, deg2, nvec);
  edge_scatter_kernel<<<gedge2, 256, 0, stream>>>(x, bufA, e2, e2 + E2, deg2, E2);
  self_scale_kernel<<<gscale, 256, 0, stream>>>(bufA, bufB, deg2, nvec);
  edge_scatter_kernel<<<gedge2, 256, 0, stream>>>(bufA, bufB, e2, e2 + E2, deg2, E2);
  gemm_wmma_kernel<<<ggemm, 256, 0, stream>>>(bufB, W2, b2, logits2, N);
  softmax_argmax_kernel<<<N, 128, 0, stream>>>(logits2, logits2, (float*)nullptr, N);
}